// PointNetSetAbstraction_43739946942747
// MI455X (gfx1250) — compile-verified
//
#include <hip/hip_runtime.h>
#include <hip/hip_bf16.h>
#include <stdint.h>

#define B_ 8
#define N_ 8192
#define S_ 2048
#define NS_ 32
#define M_ (B_ * S_ * NS_)   // 524288 rows
#define MTILES_ (M_ / 16)    // 32768 m-tiles
#define R2_ 0.04f
#define EPS_ 1e-5f

typedef __attribute__((ext_vector_type(2))) float v2f;
typedef __attribute__((ext_vector_type(8))) float v8f;
typedef __attribute__((ext_vector_type(4))) unsigned int u32x4;
typedef __attribute__((ext_vector_type(8))) int i32x8;
typedef __attribute__((ext_vector_type(4))) int i32x4;

// LDS row strides (floats): padded to break the 64-bank x 4B period.
#define LDW1_ 10   // W1p rows of 8  -> 10
#define LDW_ 66    // 64-wide rows   -> 66 (264B, 8B aligned, bank step 2)
#define LDO_ 130   // 128-wide rows  -> 130

__device__ __forceinline__ v8f wmma4(v2f a, v2f b, v8f c) {
  // D = A(16x4 f32) * B(4x16 f32) + C(16x16 f32)
  return __builtin_amdgcn_wmma_f32_16x16x4_f32(false, a, false, b, (short)0, c,
                                               false, false);
}

// ---------------------------------------------------------------------------
// new_xyz: out[b,c,s] = xyz[b,c,fps[b,s]]
// ---------------------------------------------------------------------------
__global__ void newxyz_kernel(const float* __restrict__ xyz,
                              const int* __restrict__ fps,
                              float* __restrict__ out) {
  const int i = blockIdx.x * blockDim.x + threadIdx.x;
  if (i >= B_ * 3 * S_) return;
  const int b = i / (3 * S_);
  const int r = i % (3 * S_);
  const int ch = r / S_;
  const int s = r % S_;
  out[i] = xyz[(size_t)b * 3 * N_ + (size_t)ch * N_ + fps[b * S_ + s]];
}

// ---------------------------------------------------------------------------
// Ball query: one wave32 per center; ordered first-32 selection via ballot.
// ---------------------------------------------------------------------------
__global__ __launch_bounds__(128) void ballquery_kernel(
    const float* __restrict__ xyz, const int* __restrict__ fps,
    int* __restrict__ idx) {
  const int wave = threadIdx.x >> 5;
  const int lane = threadIdx.x & 31;
  const int c = blockIdx.x * 4 + wave;  // center id in [0, B*S)
  const int b = c / S_;
  const float* xb = xyz + (size_t)b * 3 * N_;
  const int ci = fps[c];
  const float cx = xb[ci], cy = xb[N_ + ci], cz = xb[2 * N_ + ci];
  int* out = idx + (size_t)c * NS_;
  int count = 0;
  int firstn = ci;
  bool havefirst = false;
  for (int base = 0; base < N_; base += 32) {
    __builtin_prefetch(xb + base + 512, 0, 0);  // global_prefetch_b8
    const int n = base + lane;
    const float dx = xb[n] - cx;
    const float dy = xb[N_ + n] - cy;
    const float dz = xb[2 * N_ + n] - cz;
    const bool hit = (dx * dx + dy * dy + dz * dz) <= R2_;
    const unsigned long long m = __ballot(hit);  // wave32: low 32 bits
    if (!havefirst && m != 0ull) {
      havefirst = true;
      firstn = base + (__ffsll((long long)m) - 1);
    }
    if (hit) {
      const int pos = count + (int)__popcll(m & ((1ull << lane) - 1ull));
      if (pos < NS_) out[pos] = n;
    }
    count += (int)__popcll(m);
    if (count >= NS_) break;  // uniform across wave (ballot-derived)
  }
  if (count > NS_) count = NS_;
  for (int j = count + lane; j < NS_; j += 32) out[j] = firstn;  // pad w/ first hit
}

// ---------------------------------------------------------------------------
// Gather grouped features into K-padded rows: feat0[m][0..7] = {rel3, pts3, 0,0}
// ---------------------------------------------------------------------------
__global__ void gather_kernel(const float* __restrict__ xyz,
                              const float* __restrict__ pts,
                              const int* __restrict__ fps,
                              const int* __restrict__ idx,
                              float* __restrict__ feat0) {
  const int m = blockIdx.x * blockDim.x + threadIdx.x;
  if (m >= M_) return;
  const int c = m >> 5;  // center id
  const int b = c / S_;
  const int n = idx[m];
  const int ci = fps[c];
  const float* xb = xyz + (size_t)b * 3 * N_;
  const float* pb = pts + (size_t)b * 3 * N_;
  float* f = feat0 + (size_t)m * 8;
  f[0] = xb[n] - xb[ci];
  f[1] = xb[N_ + n] - xb[N_ + ci];
  f[2] = xb[2 * N_ + n] - xb[2 * N_ + ci];
  f[3] = pb[n];
  f[4] = pb[N_ + n];
  f[5] = pb[2 * N_ + n];
  f[6] = 0.f;
  f[7] = 0.f;
}

// ---------------------------------------------------------------------------
// prep: zero the 512-float stats region; pack W1 (64x6) -> W1p (64x8, zero pad)
// ---------------------------------------------------------------------------
__global__ void prep_kernel(float* __restrict__ stats,
                            const float* __restrict__ W1,
                            float* __restrict__ W1p) {
  const int t = threadIdx.x;
  if (t < 512) {
    stats[t] = 0.f;
    const int o = t >> 3, k = t & 7;
    W1p[t] = (k < 6) ? W1[o * 6 + k] : 0.f;
  }
}

// ---------------------------------------------------------------------------
// finalize: stats (sum, sumsq) -> fused BN affine: a = g*rsqrt(var+eps),
//           c = beta - mean*a (biased variance over M samples).
// ---------------------------------------------------------------------------
__global__ void finalize_kernel(const float* __restrict__ st,
                                const float* __restrict__ g,
                                const float* __restrict__ bt,
                                float* __restrict__ bn, int C) {
  const int o = threadIdx.x;
  if (o >= C) return;
  const float inv_m = 1.0f / (float)M_;
  const float mean = st[o] * inv_m;
  const float var = st[C + o] * inv_m - mean * mean;
  const float a = g[o] * rsqrtf(var + EPS_);
  bn[o] = a;
  bn[C + o] = bt[o] - mean * a;
}

// ---------------------------------------------------------------------------
// Fused MLP chain on the f32 WMMA pipe (V_WMMA_F32_16X16X4_F32).
// Weights staged once per block into bank-conflict-free padded LDS (via the
// Tensor Data Mover: 2D D# with pad_enable producing the stride-66 layout
// directly). One wave per 16-row m-tile; layers chained through per-wave LDS
// tiles with BN+ReLU fused into the A-fragment reload.
//   STAGE==1: GEMM1, accumulate layer-1 stats
//   STAGE==2: GEMM1->BN1->GEMM2, accumulate layer-2 stats
//   STAGE==3: ...->GEMM3, accumulate layer-3 stats
//   STAGE==4: full chain + BN3 + ReLU + max over 32 neighbors -> coalesced out
// Grid: 512 blocks x 128 threads (4 waves), 16 m-tiles per wave.
// ---------------------------------------------------------------------------
template <int STAGE>
__global__ __launch_bounds__(128) void stage_kernel(
    const float* __restrict__ feat0, const float* __restrict__ W1p,
    const float* __restrict__ W2, const float* __restrict__ W3,
    const float* __restrict__ bn1, const float* __restrict__ bn2,
    const float* __restrict__ bn3, float* __restrict__ sums,
    float* __restrict__ outp) {
  constexpr int WF = (STAGE == 1) ? (64 * LDW1_)
                                  : ((STAGE == 2) ? (64 * LDW1_ + 64 * LDW_)
                                                  : (64 * LDW1_ + 192 * LDW_));
  __shared__ float sW[WF];
  __shared__ float sY1[(STAGE >= 2) ? 4 : 1][16 * LDW_];
  __shared__ float sY2[(STAGE >= 3) ? 4 : 1][16 * LDW_];
  __shared__ float red[2][4][8][16];
  __shared__ float sOut[(STAGE == 4) ? 32 * LDO_ : 1];

  float* sW1 = sW;
  float* sW2 = sW + 64 * LDW1_;
  float* sW3 = sW + 64 * LDW1_ + 64 * LDW_;

  const int tid = threadIdx.x;
  const int wave = tid >> 5;
  const int lane = tid & 31;
  const int lmod = lane & 15;
  const int lhalf = lane >> 4;

  // ---- stage weights into padded LDS ----
  for (int i = tid; i < 512; i += 128) sW1[(i >> 3) * LDW1_ + (i & 7)] = W1p[i];
#if __has_builtin(__builtin_amdgcn_tensor_load_to_lds)
  if (STAGE >= 2 && wave == 0) {
    // D# group1 word0: data_size=4B, pad_enable, pad_interval=64DW, pad_amount=2DW
    const int G1W0 = (int)((2u << 16) | (1u << 20) | (5u << 22) | (1u << 25));
    const i32x4 z4 = {0, 0, 0, 0};
    const i32x8 z8 = {0, 0, 0, 0, 0, 0, 0, 0};
    {
      const unsigned long long ga = (unsigned long long)(uintptr_t)W2;
      const unsigned la = (unsigned)(uintptr_t)sW2;  // flat low 32 = LDS addr
      u32x4 g0 = {1u, la, (unsigned)(ga & 0xffffffffu),
                  (unsigned)((ga >> 32) & 0x1ffffffu) | (2u << 30)};
      i32x8 g1 = {G1W0, 64 << 16, 64 << 16, 64 << 16, 64, 64, 0, 0};
      __builtin_amdgcn_tensor_load_to_lds(g0, g1, z4, z4, z8, 0);
    }
    if (STAGE >= 3) {
      const unsigned long long ga = (unsigned long long)(uintptr_t)W3;
      const unsigned la = (unsigned)(uintptr_t)sW3;
      u32x4 g0 = {1u, la, (unsigned)(ga & 0xffffffffu),
                  (unsigned)((ga >> 32) & 0x1ffffffu) | (2u << 30)};
      i32x8 g1 = {G1W0, 64 << 16, 128 << 16, 64 << 16, 128, 64, 0, 0};
      __builtin_amdgcn_tensor_load_to_lds(g0, g1, z4, z4, z8, 0);
    }
    __builtin_amdgcn_s_wait_tensorcnt(0);
  }
#else
  if (STAGE >= 2)
    for (int i = tid; i < 64 * 64; i += 128)
      sW2[(i >> 6) * LDW_ + (i & 63)] = W2[i];
  if (STAGE >= 3)
    for (int i = tid; i < 128 * 64; i += 128)
      sW3[(i >> 6) * LDW_ + (i & 63)] = W3[i];
#endif

  float* Y1 = sY1[(STAGE >= 2) ? wave : 0];
  float* Y2 = sY2[(STAGE >= 3) ? wave : 0];

  float ssum[8], ssq[8], vmax[8];
#pragma unroll
  for (int i = 0; i < 8; ++i) {
    ssum[i] = 0.f;
    ssq[i] = 0.f;
    vmax[i] = 0.f;
  }

  const int t0 = (blockIdx.x * 4 + wave) * 16;

  for (int t = 0; t < 16; ++t) {
    const int mt = t0 + t;
    const int m0 = mt * 16;
    __syncthreads();  // staging visible / prior iteration's LDS reads complete

    // ---- layer 1: X[16x8] * W1p^T[8x64] ----
    v2f a1[2];
#pragma unroll
    for (int k = 0; k < 8; k += 4) {
      const float* pa = feat0 + (size_t)(m0 + lmod) * 8 + (k + 2 * lhalf);
      a1[k >> 2].x = pa[0];
      a1[k >> 2].y = pa[1];
    }
#pragma unroll
    for (int ot = 0; ot < 4; ++ot) {
      v8f acc = {};
#pragma unroll
      for (int k = 0; k < 8; k += 4) {
        const float* pb = sW1 + (ot * 16 + lmod) * LDW1_ + (k + 2 * lhalf);
        v2f bv;
        bv.x = pb[0];
        bv.y = pb[1];
        acc = wmma4(a1[k >> 2], bv, acc);
      }
      if (STAGE == 1) {
#pragma unroll
        for (int r = 0; r < 8; ++r) {
          const float v = acc[r];
          ssum[ot] += v;
          ssq[ot] += v * v;
        }
      } else {
#pragma unroll
        for (int r = 0; r < 8; ++r)
          Y1[(r + 8 * lhalf) * LDW_ + ot * 16 + lmod] = acc[r];
      }
    }

    if (STAGE >= 2) {
      __syncthreads();
      // ---- layer 2: relu(bn1(Y1))[16x64] * W2^T[64x64] ----
      v2f a2[16];
#pragma unroll
      for (int k = 0; k < 64; k += 4) {
        const int kc = k + 2 * lhalf;
        const float x0 = Y1[lmod * LDW_ + kc];
        const float x1 = Y1[lmod * LDW_ + kc + 1];
        a2[k >> 2].x = fmaxf(fmaf(x0, bn1[kc], bn1[64 + kc]), 0.f);
        a2[k >> 2].y = fmaxf(fmaf(x1, bn1[kc + 1], bn1[64 + kc + 1]), 0.f);
      }
#pragma unroll
      for (int ot = 0; ot < 4; ++ot) {
        v8f acc = {};
#pragma unroll
        for (int k = 0; k < 64; k += 4) {
          const float* pb = sW2 + (ot * 16 + lmod) * LDW_ + (k + 2 * lhalf);
          v2f bv;
          bv.x = pb[0];
          bv.y = pb[1];
          acc = wmma4(a2[k >> 2], bv, acc);
        }
        if (STAGE == 2) {
#pragma unroll
          for (int r = 0; r < 8; ++r) {
            const float v = acc[r];
            ssum[ot] += v;
            ssq[ot] += v * v;
          }
        } else {
#pragma unroll
          for (int r = 0; r < 8; ++r)
            Y2[(r + 8 * lhalf) * LDW_ + ot * 16 + lmod] = acc[r];
        }
      }
    }

    if (STAGE >= 3) {
      __syncthreads();
      // ---- layer 3: relu(bn2(Y2))[16x64] * W3^T[64x128] ----
      v2f a3[16];
#pragma unroll
      for (int k = 0; k < 64; k += 4) {
        const int kc = k + 2 * lhalf;
        const float x0 = Y2[lmod * LDW_ + kc];
        const float x1 = Y2[lmod * LDW_ + kc + 1];
        a3[k >> 2].x = fmaxf(fmaf(x0, bn2[kc], bn2[64 + kc]), 0.f);
        a3[k >> 2].y = fmaxf(fmaf(x1, bn2[kc + 1], bn2[64 + kc + 1]), 0.f);
      }
#pragma unroll
      for (int ot = 0; ot < 8; ++ot) {
        v8f acc = {};
#pragma unroll
        for (int k = 0; k < 64; k += 4) {
          const float* pb = sW3 + (ot * 16 + lmod) * LDW_ + (k + 2 * lhalf);
          v2f bv;
          bv.x = pb[0];
          bv.y = pb[1];
          acc = wmma4(a3[k >> 2], bv, acc);
        }
        if (STAGE == 3) {
#pragma unroll
          for (int r = 0; r < 8; ++r) {
            const float v = acc[r];
            ssum[ot] += v;
            ssq[ot] += v * v;
          }
        } else {  // STAGE == 4: BN3 + ReLU + running max over rows
          const int o = ot * 16 + lmod;
          const float sa = bn3[o];
          const float sc = bn3[128 + o];
          float mx = vmax[ot];
#pragma unroll
          for (int r = 0; r < 8; ++r)
            mx = fmaxf(mx, fmaxf(fmaf(acc[r], sa, sc), 0.f));
          vmax[ot] = mx;
        }
      }
      if (STAGE == 4 && (t & 1) == 1) {
        // finished both 16-row halves of one center's 32 neighbors
        const int cb = wave * 8 + (t >> 1);  // center within block [0,32)
#pragma unroll
        for (int ot = 0; ot < 8; ++ot) {
          float v = vmax[ot];
          v = fmaxf(v, __shfl_xor(v, 16, 32));  // merge row halves
          if (lane < 16) sOut[cb * LDO_ + ot * 16 + lmod] = v;
          vmax[ot] = 0.f;  // ReLU output >= 0, so 0 is the max identity
        }
      }
    }
  }

  // ---- stage 4: coalesced transposed writeout (32 centers x 128 ch) ----
  if (STAGE == 4) {
    __syncthreads();
    const int c0 = blockIdx.x * 32;  // first center of this block
    const int b = c0 / S_;
    const int s0 = c0 % S_;          // block stays within one batch (2048%32==0)
    for (int i = tid; i < 32 * 128; i += 128) {
      const int o = i >> 5;   // channel
      const int j = i & 31;   // center within block
      outp[(size_t)b * 128 * S_ + (size_t)o * S_ + (s0 + j)] = sOut[j * LDO_ + o];
    }
  }

  // ---- stats: wave-halves -> LDS -> cross-wave -> global atomics ----
  if (STAGE <= 3) {
    const int NOT = (STAGE == 3) ? 8 : 4;
#pragma unroll
    for (int ot = 0; ot < 8; ++ot) {
      if (ot >= NOT) break;
      ssum[ot] += __shfl_xor(ssum[ot], 16, 32);
      ssq[ot] += __shfl_xor(ssq[ot], 16, 32);
    }
    __syncthreads();
    if (lane < 16) {
      for (int ot = 0; ot < NOT; ++ot) {
        red[0][wave][ot][lmod] = ssum[ot];
        red[1][wave][ot][lmod] = ssq[ot];
      }
    }
    __syncthreads();
    const int C = NOT * 16;
    if (tid < C) {
      float s = 0.f, q = 0.f;
      for (int wv = 0; wv < 4; ++wv) {
        s += red[0][wv][tid >> 4][tid & 15];
        q += red[1][wv][tid >> 4][tid & 15];
      }
      atomicAdd(&sums[tid], s);
      atomicAdd(&sums[C + tid], q);
    }
  }
}

// ---------------------------------------------------------------------------
extern "C" void kernel_launch(void* const* d_in, const int* in_sizes, int n_in,
                              void* d_out, int out_size, void* d_ws,
                              size_t ws_size, hipStream_t stream) {
  const float* xyz = (const float*)d_in[0];
  const float* pts = (const float*)d_in[1];
  const int* fps = (const int*)d_in[2];
  const float* W1 = (const float*)d_in[3];
  const float* g1 = (const float*)d_in[4];
  const float* b1 = (const float*)d_in[5];
  const float* W2 = (const float*)d_in[6];
  const float* g2 = (const float*)d_in[7];
  const float* b2 = (const float*)d_in[8];
  const float* W3 = (const float*)d_in[9];
  const float* g3 = (const float*)d_in[10];
  const float* b3 = (const float*)d_in[11];

  float* out = (float*)d_out;
  float* new_xyz = out;                // [B,3,S]
  float* new_pts = out + B_ * 3 * S_;  // [B,128,S]

  char* w = (char*)d_ws;
  int* idx = (int*)w;        w += sizeof(int) * (size_t)M_;
  float* feat0 = (float*)w;  w += sizeof(float) * (size_t)M_ * 8;
  float* W1p = (float*)w;    w += sizeof(float) * 64 * 8;
  float* st1 = (float*)w;    w += sizeof(float) * 128;  // sum64 + sq64
  float* st2 = (float*)w;    w += sizeof(float) * 128;
  float* st3 = (float*)w;    w += sizeof(float) * 256;  // sum128 + sq128
  float* bn1 = (float*)w;    w += sizeof(float) * 128;  // a64 + c64
  float* bn2 = (float*)w;    w += sizeof(float) * 128;
  float* bn3 = (float*)w;    w += sizeof(float) * 256;

  prep_kernel<<<1, 512, 0, stream>>>(st1, W1, W1p);  // st1..st3 contiguous
  newxyz_kernel<<<(B_ * 3 * S_ + 255) / 256, 256, 0, stream>>>(xyz, fps, new_xyz);
  ballquery_kernel<<<(B_ * S_) / 4, 128, 0, stream>>>(xyz, fps, idx);
  gather_kernel<<<M_ / 256, 256, 0, stream>>>(xyz, pts, fps, idx, feat0);

  stage_kernel<1><<<512, 128, 0, stream>>>(feat0, W1p, W2, W3, bn1, bn2, bn3,
                                           st1, nullptr);
  finalize_kernel<<<1, 128, 0, stream>>>(st1, g1, b1, bn1, 64);
  stage_kernel<2><<<512, 128, 0, stream>>>(feat0, W1p, W2, W3, bn1, bn2, bn3,
                                           st2, nullptr);
  finalize_kernel<<<1, 128, 0, stream>>>(st2, g2, b2, bn2, 64);
  stage_kernel<3><<<512, 128, 0, stream>>>(feat0, W1p, W2, W3, bn1, bn2, bn3,
                                           st3, nullptr);
  finalize_kernel<<<1, 128, 0, stream>>>(st3, g3, b3, bn3, 128);
  stage_kernel<4><<<512, 128, 0, stream>>>(feat0, W1p, W2, W3, bn1, bn2, bn3,
                                           st3, new_pts);
}